// AttentionRNN_60653528154152
// MI455X (gfx1250) — compile-verified
//
#include <hip/hip_runtime.h>
#include <math.h>

// Problem constants (from reference)
#define V_  96
#define E_  64
#define H_  128
#define B_  8
#define T_  256
#define M_  (B_ * T_)   // 2048 rows

typedef __attribute__((ext_vector_type(2))) float v2f;
typedef __attribute__((ext_vector_type(8))) float v8f;

// ---------------------------------------------------------------------------
// Hardware tanh (V_TANH_F32, gfx1250 TRANS op). Prefer the builtin (compiler
// handles the TRANS-result hazard); fall back to inline asm with a V_NOP per
// ISA §7.4 (one independent op or a V_NOP must sit between a trans op and the use of its result).
// ---------------------------------------------------------------------------
__device__ __forceinline__ float htanh(float x) {
#if defined(__has_builtin) && __has_builtin(__builtin_amdgcn_tanhf)
    return __builtin_amdgcn_tanhf(x);
#else
    float r;
    asm volatile("v_tanh_f32 %0, %1\n\tv_nop" : "=v"(r) : "v"(x));
    return r;
#endif
}

// ---------------------------------------------------------------------------
// Kernel 1: pre[b,t,:] = emb[x[b,t]] @ W_ih^T + b_ih + b_hh   (WMMA f32 16x16x4)
// grid (M/16, H/16), block 32 (one wave per 16x16 tile)
// ---------------------------------------------------------------------------
__global__ void k_embed_proj(const int* __restrict__ x,
                             const float* __restrict__ emb,
                             const float* __restrict__ W_ih,
                             const float* __restrict__ b_ih,
                             const float* __restrict__ b_hh,
                             float* __restrict__ pre) {
    const int lane  = threadIdx.x;            // 0..31
    const int mt    = blockIdx.x;             // M tile
    const int nt    = blockIdx.y;             // N tile
    const int l15   = lane & 15;
    const int khalf = (lane >> 4) * 2;        // 0 or 2
    const int m     = mt * 16 + l15;          // A row for this lane
    const int n     = nt * 16 + l15;          // B col for this lane
    const int tok   = x[m];                   // token id (gathered per row)

    v8f c = {};
    #pragma unroll
    for (int k0 = 0; k0 < E_; k0 += 4) {
        const int k = k0 + khalf;
        v2f a, b;
        a.x = emb[tok * E_ + k + 0];
        a.y = emb[tok * E_ + k + 1];
        b.x = W_ih[n * E_ + k + 0];
        b.y = W_ih[n * E_ + k + 1];
        c = __builtin_amdgcn_wmma_f32_16x16x4_f32(false, a, false, b,
                                                  (short)0, c, false, false);
    }
    const float bias = b_ih[n] + b_hh[n];
    #pragma unroll
    for (int r = 0; r < 8; ++r) {
        const int mm = mt * 16 + r + 8 * (lane >> 4);
        pre[mm * H_ + n] = c[r] + bias;
    }
}

// ---------------------------------------------------------------------------
// Kernel 2a: transpose W_hh (H,H) -> Wt[j*H + i] so scan reads are coalesced
// ---------------------------------------------------------------------------
__global__ void k_transpose_whh(const float* __restrict__ W, float* __restrict__ Wt) {
    const int idx = blockIdx.x * 256 + threadIdx.x;   // 16384 total
    const int i = idx >> 7, j = idx & (H_ - 1);
    Wt[j * H_ + i] = W[idx];
}

// ---------------------------------------------------------------------------
// Kernel 2b: sequential tanh-RNN scan. One block per batch (batches are
// independent), 128 threads = 4 waves, h double-buffered in LDS, one barrier
// per timestep. Wt stays hot in L2 (64 KB). Writes rnn_out and h_last.
// ---------------------------------------------------------------------------
__global__ void k_rnn_scan(const float* __restrict__ pre,
                           const float* __restrict__ h0,
                           const float* __restrict__ Wt,
                           float* __restrict__ rnn,
                           float* __restrict__ h_last) {
    __shared__ float hbuf[2][H_];
    const int tid = threadIdx.x;     // 0..127 -> output index i
    const int b   = blockIdx.x;      // 0..7

    hbuf[0][tid] = h0[b * H_ + tid];
    __syncthreads();

    int cur = 0;
    for (int t = 0; t < T_; ++t) {
        float acc = pre[(b * T_ + t) * H_ + tid];
        const float* hc = hbuf[cur];
        #pragma unroll 8
        for (int j = 0; j < H_; ++j) {
            acc = fmaf(hc[j], Wt[j * H_ + tid], acc);   // coalesced, L2-hot
        }
        const float hn = htanh(acc);
        rnn[(b * T_ + t) * H_ + tid] = hn;
        hbuf[cur ^ 1][tid] = hn;
        __syncthreads();
        cur ^= 1;
    }
    h_last[b * H_ + tid] = hbuf[cur][tid];
}

// ---------------------------------------------------------------------------
// Kernel 3: qk = rnn @ Wq^T, kk = rnn @ Wk^T  (WMMA f32, z selects matrix)
// grid (M/16, H/16, 2), block 32
// ---------------------------------------------------------------------------
__global__ void k_proj2(const float* __restrict__ rnn,
                        const float* __restrict__ Wq,
                        const float* __restrict__ Wk,
                        float* __restrict__ qk,
                        float* __restrict__ kk) {
    const int lane  = threadIdx.x;
    const int mt    = blockIdx.x;
    const int nt    = blockIdx.y;
    const float* W  = blockIdx.z ? Wk : Wq;
    float* out      = blockIdx.z ? kk : qk;
    const int l15   = lane & 15;
    const int khalf = (lane >> 4) * 2;
    const int m     = mt * 16 + l15;
    const int n     = nt * 16 + l15;

    v8f c = {};
    #pragma unroll
    for (int k0 = 0; k0 < H_; k0 += 4) {
        const int k = k0 + khalf;
        v2f a, b;
        a.x = rnn[m * H_ + k + 0];
        a.y = rnn[m * H_ + k + 1];
        b.x = W[n * H_ + k + 0];
        b.y = W[n * H_ + k + 1];
        c = __builtin_amdgcn_wmma_f32_16x16x4_f32(false, a, false, b,
                                                  (short)0, c, false, false);
    }
    #pragma unroll
    for (int r = 0; r < 8; ++r) {
        const int mm = mt * 16 + r + 8 * (lane >> 4);
        out[mm * H_ + n] = c[r];
    }
}

// ---------------------------------------------------------------------------
// Kernel 4: causal Bahdanau attention with online softmax.
// One wave per (t,b). Lane owns 4 h-channels (float4). ~34M tanh total ->
// TRANS-bound; use hardware V_TANH_F32 (1 op/element instead of ~12).
// ---------------------------------------------------------------------------
__global__ void k_attn(const float* __restrict__ qk,
                       const float* __restrict__ kkm,
                       const float* __restrict__ rnn,
                       const float* __restrict__ v_attn,
                       float* __restrict__ ctx) {
    const int lane = threadIdx.x;   // 0..31
    const int t    = blockIdx.x;    // 0..255
    const int b    = blockIdx.y;    // 0..7

    const float4* qk4  = (const float4*)qk;
    const float4* kk4  = (const float4*)kkm;
    const float4* rnn4 = (const float4*)rnn;
    const float4  qv   = qk4[(b * T_ + t) * 32 + lane];
    const float4  va   = ((const float4*)v_attn)[lane];

    float m = -3.4e38f, l = 0.f;
    float cx0 = 0.f, cx1 = 0.f, cx2 = 0.f, cx3 = 0.f;

    for (int s = 0; s <= t; ++s) {
        const float4 kv = kk4[(b * T_ + s) * 32 + lane];
        const float t0 = htanh(qv.x + kv.x);
        const float t1 = htanh(qv.y + kv.y);
        const float t2 = htanh(qv.z + kv.z);
        const float t3 = htanh(qv.w + kv.w);
        float part = t0 * va.x;
        part = fmaf(t1, va.y, part);
        part = fmaf(t2, va.z, part);
        part = fmaf(t3, va.w, part);
        #pragma unroll
        for (int off = 16; off > 0; off >>= 1)
            part += __shfl_xor(part, off, 32);
        const float score = part;

        const float mn   = fmaxf(m, score);
        const float corr = __expf(m - mn);
        const float p    = __expf(score - mn);
        const float4 rv  = rnn4[(b * T_ + s) * 32 + lane];
        cx0 = fmaf(p, rv.x, cx0 * corr);
        cx1 = fmaf(p, rv.y, cx1 * corr);
        cx2 = fmaf(p, rv.z, cx2 * corr);
        cx3 = fmaf(p, rv.w, cx3 * corr);
        l = fmaf(l, corr, p);
        m = mn;
    }
    const float inv = 1.f / l;
    float4 o;
    o.x = cx0 * inv; o.y = cx1 * inv; o.z = cx2 * inv; o.w = cx3 * inv;
    ((float4*)ctx)[(b * T_ + t) * 32 + lane] = o;
}

// ---------------------------------------------------------------------------
// Kernel 5: logits = [rnn | ctx] @ fc_W^T + fc_b  (WMMA f32, K=256, N=96)
// grid (M/16, V/16), block 32; writes directly into d_out
// ---------------------------------------------------------------------------
__global__ void k_fc(const float* __restrict__ rnn,
                     const float* __restrict__ ctx,
                     const float* __restrict__ fcW,
                     const float* __restrict__ fcb,
                     float* __restrict__ logits) {
    const int lane  = threadIdx.x;
    const int mt    = blockIdx.x;
    const int nt    = blockIdx.y;
    const int l15   = lane & 15;
    const int khalf = (lane >> 4) * 2;
    const int m     = mt * 16 + l15;
    const int n     = nt * 16 + l15;

    v8f c = {};
    #pragma unroll
    for (int k0 = 0; k0 < 2 * H_; k0 += 4) {
        const int k = k0 + khalf;
        const float* A = (k < H_) ? (rnn + m * H_ + k) : (ctx + m * H_ + (k - H_));
        v2f a, b;
        a.x = A[0];
        a.y = A[1];
        b.x = fcW[n * (2 * H_) + k + 0];
        b.y = fcW[n * (2 * H_) + k + 1];
        c = __builtin_amdgcn_wmma_f32_16x16x4_f32(false, a, false, b,
                                                  (short)0, c, false, false);
    }
    const float bias = fcb[n];
    #pragma unroll
    for (int r = 0; r < 8; ++r) {
        const int mm = mt * 16 + r + 8 * (lane >> 4);
        logits[mm * V_ + n] = c[r] + bias;
    }
}

// ---------------------------------------------------------------------------
// Launch
// ---------------------------------------------------------------------------
extern "C" void kernel_launch(void* const* d_in, const int* in_sizes, int n_in,
                              void* d_out, int out_size, void* d_ws, size_t ws_size,
                              hipStream_t stream) {
    const int*   x      = (const int*)  d_in[0];
    const float* h0     = (const float*)d_in[1];
    const float* emb    = (const float*)d_in[2];
    const float* W_ih   = (const float*)d_in[3];
    const float* W_hh   = (const float*)d_in[4];
    const float* b_ih   = (const float*)d_in[5];
    const float* b_hh   = (const float*)d_in[6];
    const float* Wq     = (const float*)d_in[7];
    const float* Wk     = (const float*)d_in[8];
    const float* v_attn = (const float*)d_in[9];
    const float* fc_W   = (const float*)d_in[10];
    const float* fc_b   = (const float*)d_in[11];

    float* out    = (float*)d_out;
    float* logits = out;                        // (B,T,V) = 196608 floats
    float* h_last = out + M_ * V_;              // (1,B,H) = 1024 floats

    float* ws  = (float*)d_ws;
    float* pre = ws;                            // 262144 floats
    float* rnn = pre + M_ * H_;                 // 262144
    float* qk  = rnn + M_ * H_;                 // 262144
    float* kk  = qk  + M_ * H_;                 // 262144
    float* ctx = kk  + M_ * H_;                 // 262144
    float* Wt  = ctx + M_ * H_;                 // 16384

    // 1) embedding + input projection (+fused biases)
    k_embed_proj<<<dim3(M_ / 16, H_ / 16), 32, 0, stream>>>(x, emb, W_ih, b_ih, b_hh, pre);

    // 2) W_hh transpose, then batch-parallel sequential scan
    k_transpose_whh<<<dim3((H_ * H_) / 256), 256, 0, stream>>>(W_hh, Wt);
    k_rnn_scan<<<dim3(B_), H_, 0, stream>>>(pre, h0, Wt, rnn, h_last);

    // 3) query/key projections
    k_proj2<<<dim3(M_ / 16, H_ / 16, 2), 32, 0, stream>>>(rnn, Wq, Wk, qk, kk);

    // 4) causal attention + online softmax + context
    k_attn<<<dim3(T_, B_), 32, 0, stream>>>(qk, kk, rnn, v_attn, ctx);

    // 5) output FC over concat(rnn, ctx)
    k_fc<<<dim3(M_ / 16, V_ / 16), 32, 0, stream>>>(rnn, ctx, fc_W, fc_b, logits);
}